// Conv_Only_11493332484240
// MI455X (gfx1250) — compile-verified
//
#include <hip/hip_runtime.h>
#include <hip/hip_bf16.h>
#include <math.h>

#define FEAT 64
#define EPB 256   // edges staged per block

typedef __attribute__((address_space(1))) int  gint_t;
typedef __attribute__((address_space(3))) int  lint_t;

// -------- Kernel C: edge scatter  out[col,:] += w_e * x[row,:] ------------
// Defined FIRST so the disasm snippet shows the hot kernel.
// 256 threads stage EPB edge endpoints into LDS via gfx1250 async-to-LDS
// (ASYNCcnt path), compute per-edge weights cooperatively, then sweep
// 8 edges x 32 float2-lanes per pass: each edge's feature row is one
// coalesced 256B burst, edge metadata is a conflict-free LDS broadcast.
// Scatter uses native f32 global atomics (out is 25.6MB vs 192MB L2, so
// the atomic RMW traffic stays L2-resident).
__global__ void edge_scatter_kernel(const float* __restrict__ x,
                                    const int* __restrict__ rows,
                                    const int* __restrict__ cols,
                                    const float* __restrict__ d2s,
                                    const float* __restrict__ d3,
                                    float* __restrict__ out,
                                    int n_edges) {
    __shared__ int   s_row[EPB];
    __shared__ int   s_col[EPB];
    __shared__ float s_w[EPB];

    const int t = threadIdx.x;
    const long eBase = (long)blockIdx.x * EPB;
    const long eidx  = eBase + t;
    const long eclamp = (eidx < n_edges) ? eidx : (long)(n_edges - 1);

#if __has_builtin(__builtin_amdgcn_global_load_async_to_lds_b32) && __has_builtin(__builtin_amdgcn_s_wait_asynccnt)
    // gfx1250 ASYNC path: global -> LDS, tracked by ASYNCcnt
    __builtin_amdgcn_global_load_async_to_lds_b32(
        (gint_t*)(rows + eclamp), (lint_t*)(&s_row[t]), 0, 0);
    __builtin_amdgcn_global_load_async_to_lds_b32(
        (gint_t*)(cols + eclamp), (lint_t*)(&s_col[t]), 0, 0);
    __builtin_amdgcn_s_wait_asynccnt(0);
#else
    s_row[t] = rows[eclamp];
    s_col[t] = cols[eclamp];
#endif
    __syncthreads();

    // per-edge weight: w = m2 * d^e2[row] * d^e3[col]   (tail edges -> 0)
    float w = 0.0f;
    if (eidx < n_edges) w = d2s[s_row[t]] * d3[s_col[t]];
    s_w[t] = w;
    __syncthreads();

    const int esub = t >> 5;           // which of the 8 concurrent edges
    const int f2   = (t & 31) * 2;     // float2 feature offset

    #pragma unroll 4
    for (int sub = 0; sub < EPB; sub += 8) {
        int e = sub + esub;
        // prefetch a future edge's feature row toward the WGP caches
        if (sub + 32 < EPB) {
            int rpf = s_row[sub + 32 + esub];
            __builtin_prefetch(&x[(long)rpf * FEAT + f2], 0, 1);
        }
        int   row = s_row[e];
        int   col = s_col[e];
        float we  = s_w[e];
        float2 v  = *(const float2*)&x[(long)row * FEAT + f2];
        float* o  = &out[(long)col * FEAT + f2];
        atomicAdd(o + 0, we * v.x);
        atomicAdd(o + 1, we * v.y);
    }
}

// diags.pow(e) with inf -> 0 masking, matching the reference
__device__ __forceinline__ float pow_clean(float d, float e) {
    float p = powf(d, e);
    if (isinf(p)) p = 0.0f;
    return p;
}

// -------- Kernel A: per-node scalars --------------------------------------
// d2s[i] = m2 * clean(d^e2)   (m2 folded in so edge weight = d2s[row]*d3[col])
// d3[i]  = clean(d^e3)
// g2[i]  = m1*clean(d^e1) + m2*clean(d^e2)*clean(d^e3) + m3
__global__ void node_scalars_kernel(const float* __restrict__ diags,
                                    const float* __restrict__ m1p,
                                    const float* __restrict__ m2p,
                                    const float* __restrict__ m3p,
                                    const float* __restrict__ e1p,
                                    const float* __restrict__ e2p,
                                    const float* __restrict__ e3p,
                                    float* __restrict__ d2s,
                                    float* __restrict__ d3,
                                    float* __restrict__ g2,
                                    int n) {
    int i = blockIdx.x * blockDim.x + threadIdx.x;
    if (i >= n) return;
    const float m1 = m1p[0], m2 = m2p[0], m3 = m3p[0];
    const float e1 = e1p[0], e2 = e2p[0], e3 = e3p[0];
    float d  = diags[i];
    float p1 = pow_clean(d, e1);
    float p2 = pow_clean(d, e2);
    float p3 = pow_clean(d, e3);
    float w2 = m2 * p2;
    d2s[i] = w2;
    d3[i]  = p3;
    g2[i]  = fmaf(m1, p1, fmaf(w2, p3, m3));
}

// -------- Kernel B: self-loop init  out[i,:] = g2[i] * x[i,:] -------------
// float4 vectorized: one thread per 4 features (also deterministically
// overwrites the poisoned output every call).
__global__ void selfloop_init_kernel(const float* __restrict__ x,
                                     const float* __restrict__ g2,
                                     float* __restrict__ out,
                                     int n) {
    int t = blockIdx.x * blockDim.x + threadIdx.x;   // one per float4
    int total = n * (FEAT / 4);
    if (t >= total) return;
    int node = t / (FEAT / 4);
    float g = g2[node];
    float4 v = ((const float4*)x)[t];
    v.x *= g; v.y *= g; v.z *= g; v.w *= g;
    ((float4*)out)[t] = v;
}

extern "C" void kernel_launch(void* const* d_in, const int* in_sizes, int n_in,
                              void* d_out, int out_size, void* d_ws, size_t ws_size,
                              hipStream_t stream) {
    // setup_inputs() order:
    // 0 x[f32 N*64], 1 edge_index[i32 2*E], 2 edge_index_id[i32 2*N],
    // 3 diags[f32 N], 4 m1, 5 m2, 6 m3, 7 e1, 8 e2, 9 e3, 10 a
    const float* x     = (const float*)d_in[0];
    const int*   ei    = (const int*)d_in[1];
    const float* diags = (const float*)d_in[3];
    const float* m1 = (const float*)d_in[4];
    const float* m2 = (const float*)d_in[5];
    const float* m3 = (const float*)d_in[6];
    const float* e1 = (const float*)d_in[7];
    const float* e2 = (const float*)d_in[8];
    const float* e3 = (const float*)d_in[9];
    float* out = (float*)d_out;

    const int n       = in_sizes[3];
    const int n_edges = in_sizes[1] / 2;
    const int* rows = ei;             // edge_index[0]
    const int* cols = ei + n_edges;   // edge_index[1]

    // workspace: d2s[n] | d3[n] | g2[n]   (3 * 400KB)
    float* d2s = (float*)d_ws;
    float* d3  = d2s + n;
    float* g2  = d3 + n;

    node_scalars_kernel<<<(n + 255) / 256, 256, 0, stream>>>(
        diags, m1, m2, m3, e1, e2, e3, d2s, d3, g2, n);

    const int vec4 = n * (FEAT / 4);
    selfloop_init_kernel<<<(vec4 + 255) / 256, 256, 0, stream>>>(x, g2, out, n);

    edge_scatter_kernel<<<(n_edges + EPB - 1) / EPB, 256, 0, stream>>>(
        x, rows, cols, d2s, d3, out, n_edges);
}